// SimulatedClusteredAttention_26551487824101
// MI455X (gfx1250) — compile-verified
//
#include <hip/hip_runtime.h>

// ---------------- problem constants (from reference) ----------------
static constexpr int N_  = 2;
static constexpr int L_  = 4096;
static constexpr int H_  = 8;
static constexpr int E_  = 64;
static constexpr int S_  = 4096;
static constexpr int B_  = 32;     // LSH bits
static constexpr int C_  = 256;    // clusters
static constexpr int IT_ = 10;     // Lloyd iterations
static constexpr int NH_ = N_ * H_;

typedef __attribute__((ext_vector_type(2))) float v2f;
typedef __attribute__((ext_vector_type(8))) float v8f;

// ---------------- generic zero ----------------
__global__ void zero_kernel(unsigned* __restrict__ p, int n) {
    int t = blockIdx.x * blockDim.x + threadIdx.x;
    if (t < n) p[t] = 0u;
}

// ---------------- 1) LSH codes: proj = q . plane[:E] + plane[E]; bit = proj>0 ----------------
// queries: [N,L,H,E]; planes: [B, E+1]; qcodes: [NH*L] packed u32
__global__ void codes_kernel(const float* __restrict__ q,
                             const float* __restrict__ planes,
                             unsigned* __restrict__ qcodes) {
    __shared__ float sp[B_ * (E_ + 1)];
    for (int i = threadIdx.x; i < B_ * (E_ + 1); i += blockDim.x) sp[i] = planes[i];
    __syncthreads();

    int t  = blockIdx.x * blockDim.x + threadIdx.x;   // 0 .. NH*L-1
    int nh = t / L_, l = t % L_;
    int n  = nh / H_, h = nh % H_;
    const float* qr = q + (((size_t)n * L_ + l) * H_ + h) * E_;

    float acc[B_];
#pragma unroll
    for (int b = 0; b < B_; ++b) acc[b] = sp[b * (E_ + 1) + E_];   // bias
    for (int e = 0; e < E_; ++e) {
        float qe = qr[e];
#pragma unroll
        for (int b = 0; b < B_; ++b) acc[b] += qe * sp[b * (E_ + 1) + e];
    }
    unsigned code = 0u;
#pragma unroll
    for (int b = 0; b < B_; ++b) code |= (acc[b] > 0.f) ? (1u << b) : 0u;
    qcodes[t] = code;
}

// ---------------- 2) evenly spaced centroid init ----------------
__global__ void init_centroids_kernel(const unsigned* __restrict__ qcodes,
                                      const int* __restrict__ qlen,
                                      unsigned* __restrict__ ccodes) {
    int t  = blockIdx.x * blockDim.x + threadIdx.x;   // NH*C
    int nh = t / C_, c = t % C_;
    int len = qlen[nh / H_];
    int idx = (c * len) / C_;
    ccodes[t] = qcodes[nh * L_ + idx];
}

// ---------------- 3a) assignment (Hamming argmin) + LDS histogram accumulate ----------------
// grid: NH * (L/256) blocks, 256 threads; block -> one (nh, 256-query chunk)
__global__ void assign_accum_kernel(const unsigned* __restrict__ qcodes,
                                    const unsigned* __restrict__ ccodes,
                                    const int* __restrict__ qlen,
                                    int* __restrict__ bitsum,   // [NH*C*B]
                                    int* __restrict__ counts) { // [NH*C]
    __shared__ unsigned sc[C_];
    __shared__ int      sb[C_ * B_];   // 32 KB
    __shared__ int      scnt[C_];
    int tid = threadIdx.x;
    int nh  = blockIdx.x >> 4;                       // L/256 = 16 blocks per nh
    int l   = ((blockIdx.x & 15) << 8) + tid;

    sc[tid]   = ccodes[nh * C_ + tid];
    scnt[tid] = 0;
    for (int i = tid; i < C_ * B_; i += 256) sb[i] = 0;
    __syncthreads();

    unsigned qc = qcodes[nh * L_ + l];
    int best = 0, bestd = __popc(qc ^ sc[0]);
    for (int c = 1; c < C_; ++c) {
        int d = __popc(qc ^ sc[c]);
        if (d < bestd) { bestd = d; best = c; }      // first-min argmin semantics
    }
    if (l < qlen[nh / H_]) {
        atomicAdd(&scnt[best], 1);
        unsigned bits = qc;
        while (bits) {                                // add set bits only
            int b = __ffs(bits) - 1;
            bits &= bits - 1;
            atomicAdd(&sb[best * B_ + b], 1);
        }
    }
    __syncthreads();
    if (scnt[tid]) atomicAdd(&counts[nh * C_ + tid], scnt[tid]);
    for (int i = tid; i < C_ * B_; i += 256)
        if (sb[i]) atomicAdd(&bitsum[nh * C_ * B_ + i], sb[i]);
}

// ---------------- 3b) majority-vote centroid update ----------------
__global__ void update_centroids_kernel(const int* __restrict__ bitsum,
                                        const int* __restrict__ counts,
                                        unsigned* __restrict__ ccodes) {
    int t = blockIdx.x * blockDim.x + threadIdx.x;   // NH*C
    int cnt = counts[t];
    if (cnt > 0) {
        unsigned code = 0u;
        const int* bs = bitsum + t * B_;
#pragma unroll
        for (int b = 0; b < B_; ++b)
            if (2 * bs[b] > cnt) code |= (1u << b);
        ccodes[t] = code;
    }
}

// ---------------- 4) final grouping: masked per-cluster sum of query rows ----------------
// two E/2 passes keep static LDS at 32 KB + 2 KB
__global__ void final_group_kernel(const unsigned* __restrict__ qcodes,
                                   const unsigned* __restrict__ ccodes,
                                   const float* __restrict__ queries,
                                   const int* __restrict__ qlen,
                                   float* __restrict__ qsum,   // [NH*C*E]
                                   int* __restrict__ counts) { // [NH*C]
    __shared__ unsigned sc[C_];
    __shared__ float    sq[C_ * (E_ / 2)];  // 32 KB
    __shared__ int      scnt[C_];
    int tid = threadIdx.x;
    int nh  = blockIdx.x >> 4;
    int l   = ((blockIdx.x & 15) << 8) + tid;
    int n   = nh / H_, h = nh % H_;

    sc[tid]   = ccodes[nh * C_ + tid];
    scnt[tid] = 0;
    __syncthreads();

    unsigned qc = qcodes[nh * L_ + l];
    int best = 0, bestd = __popc(qc ^ sc[0]);
    for (int c = 1; c < C_; ++c) {
        int d = __popc(qc ^ sc[c]);
        if (d < bestd) { bestd = d; best = c; }
    }
    bool valid = l < qlen[n];
    const float* qr = queries + (((size_t)n * L_ + l) * H_ + h) * E_;

    for (int half = 0; half < 2; ++half) {
        for (int i = tid; i < C_ * (E_ / 2); i += 256) sq[i] = 0.f;
        __syncthreads();
        if (valid) {
            for (int e = 0; e < E_ / 2; ++e)
                atomicAdd(&sq[best * (E_ / 2) + e], qr[half * (E_ / 2) + e]);
        }
        __syncthreads();
        for (int i = tid; i < C_ * (E_ / 2); i += 256) {
            float v = sq[i];
            if (v != 0.f) {
                int c = i / (E_ / 2), e = i % (E_ / 2);
                atomicAdd(&qsum[(nh * C_ + c) * E_ + half * (E_ / 2) + e], v);
            }
        }
        __syncthreads();
    }
    if (valid) atomicAdd(&counts[nh * C_ + best], 1);
}

// ---------------- 5) masked mean: qsum /= max(count,1) ----------------
__global__ void finalize_kernel(float* __restrict__ qsum, const int* __restrict__ counts) {
    int t = blockIdx.x * blockDim.x + threadIdx.x;   // NH*C*E
    float cnt = (float)counts[t / E_];
    qsum[t] = qsum[t] / fmaxf(cnt, 1.0f);
}

// ---------------- 6) QK GEMM via fp32 WMMA 16x16x4 ----------------
// out[n,h,c,s] = sum_e Qg[nh,c,e] * keys[n,s,h,e];  one wave per 16x16 tile.
// A layout (16x4 f32): lanes 0-15 -> M=lane, VGPR0/1 = K=0/1; lanes 16-31 -> K=2/3.
// D layout: VGPR r, lanes 0-15 -> M=r, N=lane; lanes 16-31 -> M=8+r, N=lane-16.
__global__ void qk_wmma_kernel(const float* __restrict__ qg,
                               const float* __restrict__ keys,
                               float* __restrict__ out) {
    int wid  = (blockIdx.x * blockDim.x + threadIdx.x) >> 5;
    int lane = threadIdx.x & 31;

    int s_tile = wid & (S_ / 16 - 1);          // 256 tiles along S
    int c_tile = (wid >> 8) & (C_ / 16 - 1);   // 16 tiles along C
    int nh     = wid >> 12;                    // 16 (n,h) pairs
    int n = nh / H_, h = nh % H_;

    int m    = lane & 15;
    int koff = (lane >> 4) * 2;                // K sub-offset per lane half

    const float* arow = qg + ((size_t)nh * C_ + c_tile * 16 + m) * E_ + koff;
    int s = s_tile * 16 + m;                   // lane's column of B = key row s
    const float* brow = keys + (((size_t)n * S_ + s) * H_ + h) * E_ + koff;

    v8f acc = {};
#pragma unroll
    for (int e0 = 0; e0 < E_; e0 += 4) {       // 16 WMMA steps, K=4 each
        v2f a = *(const v2f*)(arow + e0);
        v2f b = *(const v2f*)(brow + e0);
        acc = __builtin_amdgcn_wmma_f32_16x16x4_f32(
            /*neg_a=*/false, a, /*neg_b=*/false, b,
            /*c_mod=*/(short)0, acc, /*reuse_a=*/false, /*reuse_b=*/false);
    }

    float* op = out + ((size_t)nh * C_ + c_tile * 16 + (lane >> 4) * 8) * S_
                    + s_tile * 16 + (lane & 15);
#pragma unroll
    for (int r = 0; r < 8; ++r) op[r * S_] = acc[r];
}

// ---------------- launch ----------------
extern "C" void kernel_launch(void* const* d_in, const int* in_sizes, int n_in,
                              void* d_out, int out_size, void* d_ws, size_t ws_size,
                              hipStream_t stream) {
    const float* queries = (const float*)d_in[0];   // [N,L,H,E]
    const float* keys    = (const float*)d_in[1];   // [N,S,H,E]
    // d_in[2] = attn_mask (unused)
    const int*   qlen    = (const int*)d_in[3];     // [N]
    const float* planes  = (const float*)d_in[4];   // [B,E+1]
    float* out = (float*)d_out;                     // [N,H,C,S]

    // workspace layout
    unsigned* qcodes = (unsigned*)d_ws;             // NH*L
    unsigned* ccodes = qcodes + NH_ * L_;           // NH*C
    int*      counts = (int*)(ccodes + NH_ * C_);   // NH*C   (contiguous with bitsum)
    int*      bitsum = counts + NH_ * C_;           // NH*C*B
    float*    qsum   = (float*)(bitsum + NH_ * C_ * B_); // NH*C*E

    // 1) LSH codes
    codes_kernel<<<NH_ * L_ / 256, 256, 0, stream>>>(queries, planes, qcodes);
    // 2) centroid init
    init_centroids_kernel<<<NH_ * C_ / 256, 256, 0, stream>>>(qcodes, qlen, ccodes);

    // 3) Lloyd iterations in Hamming space
    int histWords = NH_ * C_ * (B_ + 1);            // counts + bitsum (contiguous)
    for (int it = 0; it < IT_; ++it) {
        zero_kernel<<<(histWords + 255) / 256, 256, 0, stream>>>((unsigned*)counts, histWords);
        assign_accum_kernel<<<NH_ * (L_ / 256), 256, 0, stream>>>(qcodes, ccodes, qlen, bitsum, counts);
        update_centroids_kernel<<<NH_ * C_ / 256, 256, 0, stream>>>(bitsum, counts, ccodes);
    }

    // 4) final assignment + grouped query sums
    zero_kernel<<<(NH_ * C_ + 255) / 256, 256, 0, stream>>>((unsigned*)counts, NH_ * C_);
    zero_kernel<<<(NH_ * C_ * E_ + 255) / 256, 256, 0, stream>>>((unsigned*)qsum, NH_ * C_ * E_);
    final_group_kernel<<<NH_ * (L_ / 256), 256, 0, stream>>>(qcodes, ccodes, queries, qlen, qsum, counts);
    // 5) masked mean
    finalize_kernel<<<NH_ * C_ * E_ / 256, 256, 0, stream>>>(qsum, counts);

    // 6) centroid-key logits via fp32 WMMA
    int waves  = NH_ * (C_ / 16) * (S_ / 16);       // 65536 tiles
    int blocks = waves * 32 / 256;                  // 8 waves per block
    qk_wmma_kernel<<<blocks, 256, 0, stream>>>(qsum, keys, out);
}